// SHCodeCloud_67834713473578
// MI455X (gfx1250) — compile-verified
//
#include <hip/hip_runtime.h>
#include <hip/hip_bf16.h>

typedef __attribute__((ext_vector_type(2))) float v2f;
typedef __attribute__((ext_vector_type(8))) float v8f;

#define NPTS   65536
#define CD     64
#define KNN    8
#define SHROW  576              // CODE_DIM * SH_BASIS_DIM floats per sh row
#define WPB    4                // waves (=points) per block
#define ROWCH  144              // 16B chunks per sh row (2304 B / 16)
#define NSTAGE 36               // async b128 instrs per wave: 8*144/32

// float offset of element t of the u/G contraction inside the staged rows:
// t = k*9 + s  ->  k*576 + s
__host__ __device__ constexpr int offt(int t) { return (t / 9) * SHROW + (t % 9); }

__global__ __launch_bounds__(WPB * 32) void shcloud_fused_kernel(
    const float* __restrict__ qp,     // (NPTS,3)
    const float* __restrict__ vd,     // (NPTS,3) normalized viewdirs
    const float* __restrict__ cpos,   // (NUM_CODES,3)
    const float* __restrict__ codes,  // (NUM_CODES,64)
    const float* __restrict__ sh,     // (NUM_CODES,576) layout [d*9+s]
    const int*   __restrict__ knn,    // (NPTS,8)
    float* __restrict__ out)          // [NPTS*64 codes | NPTS*64 sh]
{
    __shared__ float stage[WPB][KNN * SHROW];   // 18432 B per wave
    __shared__ float w_lds[WPB][8];
    __shared__ float m_lds[WPB][9];
    __shared__ float u_lds[WPB][72];
    __shared__ int   j_lds[WPB][8];

    const int lane = threadIdx.x & 31;
    const int wv   = threadIdx.x >> 5;
    const int n    = blockIdx.x * WPB + wv;
    const int hi   = lane >> 4;      // lane half -> K split
    const int col  = lane & 15;      // N column within 16-wide tile
    const int dcol = 9 * col;        // float offset of column d within a row

    // ---- phase 1: neighbor indices ----
    if (lane < KNN) j_lds[wv][lane] = knn[n * KNN + lane];
    __syncthreads();

    // ---- phase 2: issue async coalesced staging of the 8 sh rows into LDS ----
    {
        unsigned dst = (unsigned)(uintptr_t)(&stage[wv][0]) + (unsigned)lane * 16u;
        int k = 0;            // row index: initial chunk = lane < 144 -> k=0
        int e = lane;         // 16B unit inside row
        #pragma unroll 4
        for (int g = 0; g < NSTAGE; ++g) {
            long j = j_lds[wv][k];
            const float* src = sh + j * SHROW + e * 4;
            asm volatile("global_load_async_to_lds_b128 %0, %1, off"
                         :: "v"(dst), "v"(src) : "memory");
            dst += 32u * 16u;               // next 512B slab
            e += 32;                        // incremental (k,e) carry update
            if (e >= ROWCH) { e -= ROWCH; k += 1; }
        }
    }

    // ---- phase 3 (overlaps staging): SH basis + IDW weights ----
    const float x = vd[n*3+0], y = vd[n*3+1], z = vd[n*3+2];
    const float xx = x*x, yy = y*y, zz = z*z;
    if (lane == 0) {
        m_lds[wv][0] = 0.28209479177387814f;
        m_lds[wv][1] = -0.4886025119029199f * y;
        m_lds[wv][2] =  0.4886025119029199f * z;
        m_lds[wv][3] = -0.4886025119029199f * x;
        m_lds[wv][4] =  1.0925484305920792f * (x*y);
        m_lds[wv][5] = -1.0925484305920792f * (y*z);
        m_lds[wv][6] =  0.31539156525252005f * (2.0f*zz - xx - yy);
        m_lds[wv][7] = -1.0925484305920792f * (x*z);
        m_lds[wv][8] =  0.5462742152960396f * (xx - yy);
    }
    if (lane < KNN) {                       // lane k owns neighbor k
        int j = j_lds[wv][lane];
        float dx = qp[n*3+0] - cpos[j*3+0];
        float dy = qp[n*3+1] - cpos[j*3+1];
        float dz = qp[n*3+2] - cpos[j*3+2];
        float d2 = dx*dx + dy*dy + dz*dz + 1e-16f;
        w_lds[wv][lane] = 1.0f / d2;        // 1/sqrt(d2)^DIST_SCALE, DIST_SCALE=2
    }
    __syncthreads();

    float wsum = 0.f;
    #pragma unroll
    for (int k = 0; k < KNN; ++k) wsum += w_lds[wv][k];
    const float winv = 1.0f / wsum;

    // ---- u[t] = w_norm[t/9] * m[t%9], t in [0,72) ----
    {
        int t0 = lane, t1 = lane + 32;
        int k0 = t0/9u, s0 = t0 - 9*k0;
        int k1 = t1/9u, s1 = t1 - 9*k1;
        u_lds[wv][t0] = w_lds[wv][k0]*winv * m_lds[wv][s0];
        u_lds[wv][t1] = w_lds[wv][k1]*winv * m_lds[wv][s1];
        if (lane < 8) {
            int t2 = lane + 64;
            int k2 = t2/9u, s2 = t2 - 9*k2;
            u_lds[wv][t2] = w_lds[wv][k2]*winv * m_lds[wv][s2];
        }
    }

    // ---- query_codes: coalesced gather, lane owns d=lane and d=lane+32 ----
    float acc0 = 0.f, acc1 = 0.f;
    #pragma unroll
    for (int k = 0; k < KNN; ++k) {
        int   j  = j_lds[wv][k];
        float wk = w_lds[wv][k] * winv;
        acc0 = fmaf(wk, codes[j*CD + lane],      acc0);
        acc1 = fmaf(wk, codes[j*CD + 32 + lane], acc1);
    }

    // ---- query_sh via WMMA f32 16x16x4: out_sh[d] = sum_t u[t]*stage[k(t)*576+9d+s(t)] ----
    v8f acc[4];
    #pragma unroll
    for (int dt = 0; dt < 4; ++dt) acc[dt] = (v8f){0.f,0.f,0.f,0.f,0.f,0.f,0.f,0.f};

    const float* up = &u_lds[wv][2 * hi];   // A base: K pair selected by lane half
    const float* bp = &stage[wv][dcol];     // B base: column d of this lane

    auto run_chunks = [&](int c0, int c1) {
        #pragma unroll
        for (int c = c0; c < c1; ++c) {
            // A: lane<16 -> K=0,1 ; lane>=16 -> K=2,3 ; same on all 16 M rows
            v2f A; A.x = up[4*c]; A.y = up[4*c + 1];
            // B: vgpr v, lane half hi -> K=2v+hi -> t = 4c + 2v + hi
            // offt(t) is a compile-time constant per (c, hi); one cndmask each
            int off0 = hi ? offt(4*c + 1) : offt(4*c);
            int off1 = hi ? offt(4*c + 3) : offt(4*c + 2);
            const float* b0 = bp + off0;
            const float* b1 = bp + off1;
            #pragma unroll
            for (int dt = 0; dt < 4; ++dt) {
                v2f B;
                B.x = b0[144 * dt];          // d = 16*dt + col, row stride 9*16
                B.y = b1[144 * dt];
                acc[dt] = __builtin_amdgcn_wmma_f32_16x16x4_f32(
                    false, A, false, B, (short)0, acc[dt], false, false);
            }
        }
    };

    // chunks 0..8 need rows k<=3 == first 18 staging instrs (chunks < 576)
    asm volatile("s_wait_asynccnt 18" ::: "memory");
    run_chunks(0, 9);
    asm volatile("s_wait_asynccnt 0" ::: "memory");
    run_chunks(9, 18);

    // ---- writeback ----
    out[(size_t)n*CD + lane]      = acc0;
    out[(size_t)n*CD + 32 + lane] = acc1;
    if (lane < 16) {
        float* osh = out + (size_t)NPTS*CD + (size_t)n*CD;
        osh[lane]      = acc[0][0];   // all C rows identical; row 0, col = lane
        osh[16 + lane] = acc[1][0];
        osh[32 + lane] = acc[2][0];
        osh[48 + lane] = acc[3][0];
    }
}

extern "C" void kernel_launch(void* const* d_in, const int* in_sizes, int n_in,
                              void* d_out, int out_size, void* d_ws, size_t ws_size,
                              hipStream_t stream) {
    const float* qp    = (const float*)d_in[0];  // query_points (1,N,3)
    const float* vd    = (const float*)d_in[1];  // viewdirs (N,3)
    const float* cpos  = (const float*)d_in[2];  // codes_position (1,8192,3)
    const float* codes = (const float*)d_in[3];  // codes (1,8192,64)
    const float* sh    = (const float*)d_in[4];  // sh_codes (1,8192,576)
    // d_in[5] = indices (batch selector, always 0) -- unused
    const int*   knn   = (const int*)d_in[6];    // knn_idx (N,8)
    float* out = (float*)d_out;

    dim3 grid(NPTS / WPB);    // 16384 blocks
    dim3 block(WPB * 32);     // 4 wave32 = 4 points per block
    shcloud_fused_kernel<<<grid, block, 0, stream>>>(qp, vd, cpos, codes, sh, knn, out);
}